// TemporalEmbedding_3839700763037
// MI455X (gfx1250) — compile-verified
//
#include <hip/hip_runtime.h>

typedef __attribute__((ext_vector_type(2))) float v2f;
typedef __attribute__((ext_vector_type(8))) float v8f;
typedef __attribute__((ext_vector_type(4))) float f4;

#define DM 512           // d_model
#define PSEQ 2048        // sequence length P
#define NTOK (32 * 2048) // B * P tokens per stream

// ---------------------------------------------------------------------------
// Tiny fp32 GEMM:  T[M,512] = A[M,512] @ Wb[512,512]  (+ optional bias)
// One wave (32 lanes) per 16x16 output tile, V_WMMA_F32_16X16X4_F32 over K.
// VGPR layouts per CDNA5 ISA 7.12.2:
//   A 16x4 : lanes hold M=lane&15; VGPR0=K{0|2}, VGPR1=K{1|3} (half selects +2)
//   B 4x16 : lanes hold N=lane&15; VGPR0 = row K{0|2}, VGPR1 = row K{1|3}
//   D 16x16: VGPR v holds row (8*half + v), col (lane&15)
// M-tail handling: clamp the A row POINTER only (in-bounds address); garbage
// accumulator rows m>=M are never stored, and A row m feeds only D row m, so
// no zero-masking (and no EXEC churn) is needed. EXEC is all-1s throughout.
// ---------------------------------------------------------------------------
__global__ __launch_bounds__(32)
void wmma_table_gemm(const float* __restrict__ A, int M,
                     const float* __restrict__ Wb,
                     const float* __restrict__ bias, int has_bias,
                     float* __restrict__ T) {
  const int lane  = threadIdx.x;
  const int nTile = blockIdx.x << 4;
  const int mTile = blockIdx.y << 4;
  const int half  = lane >> 4;       // 0 or 1
  const int idx   = lane & 15;
  const int koff  = half << 1;       // 0 or 2

  const int m  = mTile + idx;
  const int n  = nTile + idx;
  const int mc = (m < M) ? m : (M - 1);          // clamp address, not data
  const float* __restrict__ Arow = A + (size_t)mc * DM + koff;
  const float* __restrict__ Bcol = Wb + (size_t)koff * DM + n;

  v8f acc = {};
  for (int k0 = 0; k0 < DM; k0 += 4) {
    // A pair {K=koff, K=koff+1}: contiguous, 8B-aligned -> one b64 load
    const v2f av = *(const v2f*)(Arow + k0);
    v2f bv;
    bv.x = Bcol[(size_t)k0 * DM];                // row k0+koff
    bv.y = Bcol[(size_t)k0 * DM + DM];           // row k0+koff+1
    acc = __builtin_amdgcn_wmma_f32_16x16x4_f32(
        /*neg_a=*/false, av, /*neg_b=*/false, bv,
        /*c_mod=*/(short)0, acc, /*reuse_a=*/false, /*reuse_b=*/false);
  }

  const float bb = has_bias ? bias[n] : 0.0f;
#pragma unroll
  for (int v = 0; v < 8; ++v) {
    const int row = mTile + (half << 3) + v;
    if (row < M) T[(size_t)row * DM + n] = acc[v] + bb;
  }
}

// ---------------------------------------------------------------------------
// Streaming gather-sum: out[tok, :] = T0[i0] + T1[i1] + T2[i2] + PE'[l]
// (bias already folded into PE'). float4 per thread -> b128 loads/stores.
// Tables (~4.9 MB) are L2-resident; this pass is HBM-write-bound:
// ~256 MB stores + ~16 MB L2-hit reads / 23.3 TB/s  ->  ~11.6 us floor.
// ---------------------------------------------------------------------------
__global__ __launch_bounds__(256)
void gather_sum(const int* __restrict__ i0, const int* __restrict__ i1,
                const int* __restrict__ i2,
                const f4* __restrict__ T0, const f4* __restrict__ T1,
                const f4* __restrict__ T2, const f4* __restrict__ PEb,
                f4* __restrict__ out) {
  const int gid = blockIdx.x * 256 + threadIdx.x;  // over NTOK * 128 float4s
  const int dv  = gid & 127;                       // which float4 of d_model
  const int tok = gid >> 7;                        // token index
  const int l   = tok & (PSEQ - 1);                // position within sequence

  const f4 a = T0[(size_t)i0[tok] * 128 + dv];
  const f4 b = T1[(size_t)i1[tok] * 128 + dv];
  const f4 c = T2[(size_t)i2[tok] * 128 + dv];
  const f4 p = PEb[(size_t)l * 128 + dv];
  out[gid] = a + b + c + p;
}

extern "C" void kernel_launch(void* const* d_in, const int* in_sizes, int n_in,
                              void* d_out, int out_size, void* d_ws, size_t ws_size,
                              hipStream_t stream) {
  (void)in_sizes; (void)n_in; (void)out_size; (void)ws_size;

  const int*   in0  = (const int*)d_in[0];
  const int*   tg0  = (const int*)d_in[1];
  const int*   in1  = (const int*)d_in[2];
  const int*   tg1  = (const int*)d_in[3];
  const int*   in2  = (const int*)d_in[4];
  const int*   tg2  = (const int*)d_in[5];
  const float* emb0 = (const float*)d_in[6];   // [7,   512]
  const float* emb1 = (const float*)d_in[7];   // [24,  512]
  const float* emb2 = (const float*)d_in[8];   // [288, 512]
  const float* pe   = (const float*)d_in[9];   // [4096,512] (use first 2048 rows)
  const float* W    = (const float*)d_in[10];  // [2048,512] row-major
  const float* bias = (const float*)d_in[11];  // [512]

  // Workspace layout (16B-aligned offsets)
  char*  ws  = (char*)d_ws;
  float* T0  = (float*)(ws + 0);        //    7*512*4 = 14336 B
  float* T1  = (float*)(ws + 16384);    //   24*512*4 = 49152 B
  float* T2  = (float*)(ws + 65536);    //  288*512*4 = 589824 B
  float* PEb = (float*)(ws + 655360);   // 2048*512*4 = 4 MiB (bias folded in)

  // W blocks: rows [s*512, (s+1)*512) multiply segment s of the concat
  const size_t WBLK = (size_t)DM * DM;  // 262144 floats

  const dim3 blk(32);
  wmma_table_gemm<<<dim3(32,   1), blk, 0, stream>>>(emb0,    7, W + 0 * WBLK, bias, 0, T0);
  wmma_table_gemm<<<dim3(32,   2), blk, 0, stream>>>(emb1,   24, W + 1 * WBLK, bias, 0, T1);
  wmma_table_gemm<<<dim3(32,  18), blk, 0, stream>>>(emb2,  288, W + 2 * WBLK, bias, 0, T2);
  wmma_table_gemm<<<dim3(32, 128), blk, 0, stream>>>(pe,   PSEQ, W + 3 * WBLK, bias, 1, PEb);

  // Streaming pass: 2 streams x 32*2048 tokens x 128 float4
  const int blocks = (NTOK * 128) / 256;  // 32768
  f4* out0 = (f4*)d_out;
  f4* out1 = out0 + (size_t)NTOK * 128;
  gather_sum<<<blocks, 256, 0, stream>>>(in0, in1, in2,
                                         (const f4*)T0, (const f4*)T1,
                                         (const f4*)T2, (const f4*)PEb, out0);
  gather_sum<<<blocks, 256, 0, stream>>>(tg0, tg1, tg2,
                                         (const f4*)T0, (const f4*)T1,
                                         (const f4*)T2, (const f4*)PEb, out1);
}